// SplitAndMix_24086176596208
// MI455X (gfx1250) — compile-verified
//
#include <hip/hip_runtime.h>
#include <stdint.h>

// SplitAndMix on MI455X (gfx1250): pure streaming gather, HBM-bandwidth bound.
// 402 MB total traffic -> ~17us floor at 23.3 TB/s. Strategy:
//   - one wave32 per token; each lane moves 16 B, so every (token, track)
//     chunk is one contiguous, fully-coalesced 512 B transfer
//   - async global->LDS copies (GLOBAL_LOAD_ASYNC_TO_LDS_B128, ASYNCcnt) keep
//     12 x 512 B in flight per wave with no data-VGPR pressure
//   - non-temporal stores: 384 MB streamed once, don't thrash the 192 MB L2
//   - wave-uniform index math forced onto the scalar unit via readfirstlane

typedef float v4f __attribute__((ext_vector_type(4)));
typedef int   v4i __attribute__((ext_vector_type(4)));

// Address-space-qualified v4i pointers, matching the builtin's signature.
typedef __attribute__((address_space(1))) v4i gv4i;  // global
typedef __attribute__((address_space(3))) v4i lv4i;  // LDS

#define N_TRACKS 12
#define TRACK_SIZE 128
#define EMB 1536               // N_TRACKS * TRACK_SIZE
#define WAVES_PER_BLOCK 8
#define LANES 32

#ifndef __has_builtin
#define __has_builtin(x) 0
#endif

#if __has_builtin(__builtin_amdgcn_global_load_async_to_lds_b128) && \
    __has_builtin(__builtin_amdgcn_s_wait_asynccnt)
#define USE_ASYNC_LDS 1
#else
#define USE_ASYNC_LDS 0
#endif

__global__ __launch_bounds__(256) void splitmix_kernel(
    const float* __restrict__ data,
    const int*   __restrict__ lengths,
    float*       __restrict__ out,
    int total, int n_seq)
{
    const int lane = threadIdx.x & (LANES - 1);
    const int wave = threadIdx.x >> 5;

    // Token index is wave-uniform; pin it to an SGPR so the sequence search
    // and row math compile to SALU (no exec-mask waterfall loop).
    const int tok = __builtin_amdgcn_readfirstlane(
        (int)(blockIdx.x * WAVES_PER_BLOCK + wave));
    if (tok >= total) return;   // uniform branch

    // Branchless scan over the (tiny) lengths array: no early break, so the
    // compiler emits straight-line s_load + s_cselect instead of a loop with
    // divergence bookkeeping.
    int segStart = 0, L = 1, acc = 0;
    for (int s = 0; s < n_seq; ++s) {
        int  len  = lengths[s];
        int  next = acc + len;
        bool in   = (tok >= acc) & (tok < next);
        segStart  = in ? acc : segStart;
        L         = in ? len : L;
        acc       = next;
    }
    const int pos = tok - segStart;

    // Source row per track: segStart + (pos + shift_i) mod L, shift_i = 2^(i-1).
    int srcRow[N_TRACKS];
#pragma unroll
    for (int i = 0; i < N_TRACKS; ++i) {
        int shift = (i == 0) ? 0 : (1 << (i - 1));
        int j = pos + shift;
        while (j >= L) j -= L;   // shift_max = 1024 < L_min = 2048: <=1 iter, uniform
        srcRow[i] = segStart + j;
    }

    v4f* __restrict__ dst4 = (v4f*)(out + (size_t)tok * EMB);

#if USE_ASYNC_LDS
    // ---- CDNA5 async global->LDS pipeline (ASYNCcnt) -----------------------
    __shared__ v4f stage[WAVES_PER_BLOCK][N_TRACKS][LANES];  // 48 KB / block

    // Issue all 12 gathers; each lane pulls 16 B straight into LDS, no VGPRs.
#pragma unroll
    for (int i = 0; i < N_TRACKS; ++i) {
        const float* gp = data + (size_t)srcRow[i] * EMB + i * TRACK_SIZE + lane * 4;
        __builtin_amdgcn_global_load_async_to_lds_b128(
            (gv4i*)gp, (lv4i*)&stage[wave][i][lane], /*offset=*/0, /*cpol=*/0);
    }

    // Drain in order: async loads complete in order, so waiting for
    // asynccnt <= (11 - i) guarantees track i has landed in LDS.
#define CONSUME(I)                                                        \
    do {                                                                  \
        __builtin_amdgcn_s_wait_asynccnt(N_TRACKS - 1 - (I));             \
        asm volatile("" ::: "memory");                                    \
        v4f v = stage[wave][(I)][lane];                                   \
        __builtin_nontemporal_store(v, dst4 + (I) * LANES + lane);        \
    } while (0)

    CONSUME(0);  CONSUME(1);  CONSUME(2);  CONSUME(3);
    CONSUME(4);  CONSUME(5);  CONSUME(6);  CONSUME(7);
    CONSUME(8);  CONSUME(9);  CONSUME(10); CONSUME(11);
#undef CONSUME

#else
    // ---- Fallback: direct NT b128 loads into registers, then NT stores ----
    v4f v[N_TRACKS];
#pragma unroll
    for (int i = 0; i < N_TRACKS; ++i) {
        const v4f* p = (const v4f*)(data + (size_t)srcRow[i] * EMB + i * TRACK_SIZE) + lane;
        v[i] = __builtin_nontemporal_load(p);
    }
#pragma unroll
    for (int i = 0; i < N_TRACKS; ++i)
        __builtin_nontemporal_store(v[i], dst4 + i * LANES + lane);
#endif
}

extern "C" void kernel_launch(void* const* d_in, const int* in_sizes, int n_in,
                              void* d_out, int out_size, void* d_ws, size_t ws_size,
                              hipStream_t stream) {
    const float* data    = (const float*)d_in[0];
    const int*   lengths = (const int*)d_in[1];
    float*       out     = (float*)d_out;

    const int total = in_sizes[0] / EMB;   // 32768 tokens
    const int n_seq = in_sizes[1];         // 8 sequences

    const int blocks = (total + WAVES_PER_BLOCK - 1) / WAVES_PER_BLOCK;
    splitmix_kernel<<<blocks, 256, 0, stream>>>(data, lengths, out, total, n_seq);
}